// PTSeg_1623497638708
// MI455X (gfx1250) — compile-verified
//
#include <hip/hip_runtime.h>
#include <math.h>

typedef __attribute__((ext_vector_type(2))) float v2f;
typedef __attribute__((ext_vector_type(8))) float v8f;

#define NPTS   100000
#define CCH    64
#define NSAMP  16
#define CS     8        // C / share_planes
#define BN_EPS 1e-5f

// ---------------------------------------------------------------------------
// Kernel A: fused Q/K/V projection  (q = x@Wq+bq, k = x@Wk+bk, v = x@Wv+bv)
// One wave32 computes a 16-row x 64-col tile of all three outputs with
// V_WMMA_F32_16X16X4_F32 (full fp32 precision, matches the fp32 reference).
//
// A (16x4 f32) layout: lanes 0-15 -> M=lane, VGPR0=K0,VGPR1=K1;
//                      lanes 16-31 -> M=lane-16, VGPR0=K2,VGPR1=K3.
// B (4x16 f32) layout (symmetric): lanes 0-15 -> N=lane, K0/K1;
//                                  lanes 16-31 -> N=lane-16, K2/K3.
// D (16x16 f32): VGPR r, lanes 0-15 -> (M=r, N=lane); lanes 16-31 -> (M=r+8).
// ---------------------------------------------------------------------------
__global__ __launch_bounds__(256)
void qkv_wmma_kernel(const float* __restrict__ x,
                     const float* __restrict__ Wq, const float* __restrict__ bq,
                     const float* __restrict__ Wk, const float* __restrict__ bk,
                     const float* __restrict__ Wv, const float* __restrict__ bv,
                     float* __restrict__ q, float* __restrict__ kOut,
                     float* __restrict__ vOut, int ntiles) {
  const int wave = blockIdx.x * (blockDim.x >> 5) + (threadIdx.x >> 5);
  if (wave >= ntiles) return;             // wave-uniform -> EXEC stays all-ones
  const int lane = threadIdx.x & 31;
  const int l15  = lane & 15;
  const int hi   = lane >> 4;             // 0: K pair {0,1}, 1: K pair {2,3}
  const int row  = wave * 16 + l15;

  const float* W[3]    = {Wq, Wk, Wv};
  const float* bias[3] = {bq, bk, bv};
  float* out[3]        = {q, kOut, vOut};

  v8f acc[3][4];
#pragma unroll
  for (int m = 0; m < 3; ++m)
#pragma unroll
    for (int t = 0; t < 4; ++t)
      acc[m][t] = (v8f){0.f, 0.f, 0.f, 0.f, 0.f, 0.f, 0.f, 0.f};

#pragma unroll 4
  for (int kk = 0; kk < CCH; kk += 4) {
    // A fragment: x[row, kk + 2*hi .. kk + 2*hi + 1]  (8B-aligned float2)
    v2f a = *(const v2f*)(x + row * CCH + kk + 2 * hi);
#pragma unroll
    for (int m = 0; m < 3; ++m) {
      const float* Wm = W[m];
#pragma unroll
      for (int t = 0; t < 4; ++t) {
        const int col = t * 16 + l15;
        v2f b;
        b.x = Wm[(kk + 2 * hi + 0) * CCH + col];
        b.y = Wm[(kk + 2 * hi + 1) * CCH + col];
        acc[m][t] = __builtin_amdgcn_wmma_f32_16x16x4_f32(
            false, a, false, b, (short)0, acc[m][t], false, false);
      }
    }
  }

#pragma unroll
  for (int m = 0; m < 3; ++m)
#pragma unroll
    for (int t = 0; t < 4; ++t) {
      const int col  = t * 16 + l15;
      const float bc = bias[m][col];
#pragma unroll
      for (int r = 0; r < 8; ++r) {
        const int orow = wave * 16 + r + hi * 8;
        out[m][orow * CCH + col] = acc[m][t][r] + bc;
      }
    }
}

// ---------------------------------------------------------------------------
// Kernel B: gather + positional MLP + vector-attention weights + softmax +
// grouped weighted sum. One wave32 per point; lane owns channels {l, l+32}.
// ---------------------------------------------------------------------------
__global__ __launch_bounds__(128)
void attn_kernel(const float* __restrict__ p, const int* __restrict__ idx,
                 const float* __restrict__ q, const float* __restrict__ kx,
                 const float* __restrict__ vx,
                 const float* __restrict__ Wp1, const float* __restrict__ bp1,
                 const float* __restrict__ gp,  const float* __restrict__ bp,
                 const float* __restrict__ mp,  const float* __restrict__ vp,
                 const float* __restrict__ Wp2, const float* __restrict__ bp2,
                 const float* __restrict__ gw1, const float* __restrict__ bw1b,
                 const float* __restrict__ mw1, const float* __restrict__ vw1,
                 const float* __restrict__ Ww1, const float* __restrict__ bw1,
                 const float* __restrict__ gw2, const float* __restrict__ bw2b,
                 const float* __restrict__ mw2, const float* __restrict__ vw2,
                 const float* __restrict__ Ww2, const float* __restrict__ bw2,
                 float* __restrict__ out) {
  __shared__ float lds_vs[4][NSAMP][CCH];  // (v_gathered + pos_encoding)
  __shared__ float lds_w[4][NSAMP][CS];    // attention logits / weights

  const int wid  = threadIdx.x >> 5;
  const int lane = threadIdx.x & 31;
  const int n    = blockIdx.x * 4 + wid;   // NPTS % 4 == 0
  const int c0   = lane, c1 = lane + 32;
  const int jl   = lane & 7;

  const float px = p[n * 3 + 0], py = p[n * 3 + 1], pz = p[n * 3 + 2];
  const float q0 = q[(long)n * CCH + c0], q1 = q[(long)n * CCH + c1];

  // fold BN(3) of linear_p:  bn(x) = s*x + t
  float sp[3], tp[3];
#pragma unroll
  for (int d = 0; d < 3; ++d) {
    const float s = gp[d] * rsqrtf(vp[d] + BN_EPS);
    sp[d] = s; tp[d] = bp[d] - mp[d] * s;
  }
  // fold BN(C) (first BN of linear_w) for this lane's two channels
  const float s1_0 = gw1[c0] * rsqrtf(vw1[c0] + BN_EPS);
  const float t1_0 = bw1b[c0] - mw1[c0] * s1_0;
  const float s1_1 = gw1[c1] * rsqrtf(vw1[c1] + BN_EPS);
  const float t1_1 = bw1b[c1] - mw1[c1] * s1_1;
  // Wp2 columns + bias for this lane's channels
  float pc0[3], pc1[3];
#pragma unroll
  for (int d = 0; d < 3; ++d) { pc0[d] = Wp2[d * CCH + c0]; pc1[d] = Wp2[d * CCH + c1]; }
  const float bp2_0 = bp2[c0], bp2_1 = bp2[c1];
  // Ww1 rows for this lane's channels
  float w1r0[CS], w1r1[CS];
#pragma unroll
  for (int j = 0; j < CS; ++j) { w1r0[j] = Ww1[c0 * CS + j]; w1r1[j] = Ww1[c1 * CS + j]; }
  // stage-2 params (replicated per lane): bias, folded BN(C//S)
  float bw1v[CS], s2[CS], t2[CS];
#pragma unroll
  for (int j = 0; j < CS; ++j) {
    bw1v[j] = bw1[j];
    const float s = gw2[j] * rsqrtf(vw2[j] + BN_EPS);
    s2[j] = s; t2[j] = bw2b[j] - mw2[j] * s;
  }
  // Ww2 column for lane's output channel (lanes 0-7 produce the 8 logits)
  float w2col[CS];
#pragma unroll
  for (int j = 0; j < CS; ++j) w2col[j] = Ww2[j * CS + jl];
  const float bw2l = bw2[jl];
  // Wp1 (9) + bp1 (3), replicated
  float wp1[9], bp1v[3];
#pragma unroll
  for (int i = 0; i < 9; ++i) wp1[i] = Wp1[i];
#pragma unroll
  for (int d = 0; d < 3; ++d) bp1v[d] = bp1[d];

  for (int ns = 0; ns < NSAMP; ++ns) {
    const int j = idx[n * NSAMP + ns];
    if (ns + 1 < NSAMP) {                 // hide gather latency: global_prefetch_b8
      const int jn = idx[n * NSAMP + ns + 1];
      __builtin_prefetch(kx + (long)jn * CCH + c0, 0, 1);
      __builtin_prefetch(vx + (long)jn * CCH + c0, 0, 1);
    }
    // relative position + linear_p front (3->3, BN, ReLU)
    const float prx = p[j * 3 + 0] - px;
    const float pry = p[j * 3 + 1] - py;
    const float prz = p[j * 3 + 2] - pz;
    float t3[3];
#pragma unroll
    for (int d = 0; d < 3; ++d) {
      float u = prx * wp1[0 * 3 + d] + pry * wp1[1 * 3 + d] + prz * wp1[2 * 3 + d] + bp1v[d];
      t3[d] = fmaxf(sp[d] * u + tp[d], 0.f);
    }
    // positional encoding (3 -> C), this lane's two channels
    const float pe0 = t3[0] * pc0[0] + t3[1] * pc0[1] + t3[2] * pc0[2] + bp2_0;
    const float pe1 = t3[0] * pc1[0] + t3[1] * pc1[1] + t3[2] * pc1[2] + bp2_1;
    // coalesced gathers of k/v rows
    const float kg0 = kx[(long)j * CCH + c0], kg1 = kx[(long)j * CCH + c1];
    const float vg0 = vx[(long)j * CCH + c0], vg1 = vx[(long)j * CCH + c1];
    lds_vs[wid][ns][c0] = vg0 + pe0;
    lds_vs[wid][ns][c1] = vg1 + pe1;
    // w = k_g - q + pe  -> BN -> ReLU
    const float r0 = fmaxf(s1_0 * (kg0 - q0 + pe0) + t1_0, 0.f);
    const float r1 = fmaxf(s1_1 * (kg1 - q1 + pe1) + t1_1, 0.f);
    // 64 -> 8 contraction: per-lane partials + cross-lane tree reduction
    float acc8[CS];
#pragma unroll
    for (int jj = 0; jj < CS; ++jj) acc8[jj] = r0 * w1r0[jj] + r1 * w1r1[jj];
#pragma unroll
    for (int off = 16; off >= 1; off >>= 1)
#pragma unroll
      for (int jj = 0; jj < CS; ++jj) acc8[jj] += __shfl_xor(acc8[jj], off, 32);
    // + bias -> BN -> ReLU (replicated on all lanes)
    float h[CS];
#pragma unroll
    for (int jj = 0; jj < CS; ++jj)
      h[jj] = fmaxf(s2[jj] * (acc8[jj] + bw1v[jj]) + t2[jj], 0.f);
    // 8 -> 8: lanes 0-7 each produce one logit
    float o = bw2l;
#pragma unroll
    for (int jj = 0; jj < CS; ++jj) o += h[jj] * w2col[jj];
    if (lane < CS) lds_w[wid][ns][lane] = o;
  }
  __syncthreads();

  // per-channel softmax over the 16 neighbors (lanes 0-7, one channel each)
  if (lane < CS) {
    float m = -3.4e38f;
#pragma unroll
    for (int ns = 0; ns < NSAMP; ++ns) m = fmaxf(m, lds_w[wid][ns][lane]);
    float e[NSAMP], s = 0.f;
#pragma unroll
    for (int ns = 0; ns < NSAMP; ++ns) { e[ns] = __expf(lds_w[wid][ns][lane] - m); s += e[ns]; }
    const float inv = 1.f / s;
#pragma unroll
    for (int ns = 0; ns < NSAMP; ++ns) lds_w[wid][ns][lane] = e[ns] * inv;
  }
  __syncthreads();

  // grouped weighted sum: out[c] = sum_ns (v+pe)[ns][c] * w[ns][c%8]
  float o0 = 0.f, o1 = 0.f;
#pragma unroll
  for (int ns = 0; ns < NSAMP; ++ns) {
    const float wch = lds_w[wid][ns][jl];
    o0 += lds_vs[wid][ns][c0] * wch;
    o1 += lds_vs[wid][ns][c1] * wch;
  }
  out[(long)n * CCH + c0] = o0;
  out[(long)n * CCH + c1] = o1;
}

// ---------------------------------------------------------------------------
extern "C" void kernel_launch(void* const* d_in, const int* in_sizes, int n_in,
                              void* d_out, int out_size, void* d_ws, size_t ws_size,
                              hipStream_t stream) {
  const float* p    = (const float*)d_in[0];
  const float* x    = (const float*)d_in[1];
  const int*   idx  = (const int*)d_in[2];     // jnp "int64" is int32 (x64 off)
  const float* Wq   = (const float*)d_in[3];
  const float* bq   = (const float*)d_in[4];
  const float* Wk   = (const float*)d_in[5];
  const float* bk   = (const float*)d_in[6];
  const float* Wv   = (const float*)d_in[7];
  const float* bv   = (const float*)d_in[8];
  const float* Wp1  = (const float*)d_in[9];
  const float* bp1  = (const float*)d_in[10];
  const float* gp   = (const float*)d_in[11];
  const float* bp   = (const float*)d_in[12];
  const float* mp   = (const float*)d_in[13];
  const float* vp   = (const float*)d_in[14];
  const float* Wp2  = (const float*)d_in[15];
  const float* bp2  = (const float*)d_in[16];
  const float* gw1  = (const float*)d_in[17];
  const float* bw1b = (const float*)d_in[18];
  const float* mw1  = (const float*)d_in[19];
  const float* vw1  = (const float*)d_in[20];
  const float* Ww1  = (const float*)d_in[21];
  const float* bw1  = (const float*)d_in[22];
  const float* gw2  = (const float*)d_in[23];
  const float* bw2b = (const float*)d_in[24];
  const float* mw2  = (const float*)d_in[25];
  const float* vw2  = (const float*)d_in[26];
  const float* Ww2  = (const float*)d_in[27];
  const float* bw2  = (const float*)d_in[28];

  float* q = (float*)d_ws;                 // q/k/v scratch: 3 * N * C floats
  float* k = q + (size_t)NPTS * CCH;
  float* v = k + (size_t)NPTS * CCH;

  const int ntiles = NPTS / 16;            // 6250 row tiles of 16
  qkv_wmma_kernel<<<(ntiles + 7) / 8, 256, 0, stream>>>(
      x, Wq, bq, Wk, bk, Wv, bv, q, k, v, ntiles);

  attn_kernel<<<NPTS / 4, 128, 0, stream>>>(
      p, idx, q, k, v,
      Wp1, bp1, gp, bp, mp, vp, Wp2, bp2,
      gw1, bw1b, mw1, vw1, Ww1, bw1,
      gw2, bw2b, mw2, vw2, Ww2, bw2,
      (float*)d_out);
}